// Edge_Gate_Convolution_13194139533628
// MI455X (gfx1250) — compile-verified
//
#include <hip/hip_runtime.h>
#include <hip/hip_bf16.h>
#include <math.h>

// CDNA5 (gfx1250) wave32 WMMA implementation of edge-gated graph conv.
// All GEMMs use V_WMMA_F32_16X16X4_F32 (full fp32): the pipeline is
// memory-bound (~13 FLOP/B << fp32-WMMA roofline crossover), so fp32
// matrix cores are free and bit-faithful.

typedef __attribute__((ext_vector_type(2))) float v2f;
typedef __attribute__((ext_vector_type(8))) float v8f;

#define FDIM 128
#define KSTEPS 32          // 128 / 4
#define TILE_ROWS 128      // rows per block (8 waves x 16 rows)
#define NTHREADS 256

__device__ __forceinline__ float silu_f(float x) {
    return x / (1.0f + __expf(-x));
}

// ---------------------------------------------------------------------------
// Y[rows,128] = X[rows,128] @ W[128,128] + bias
// A layout (16x4 f32): lanes 0-15 hold K=0(lo),1(hi); lanes 16-31 K=2,3.
// B layout (4x16):     VGPR pair row-striped; packed W pairs (k,k+1) in LDS
//                      so each lane's B operand is one ds_load_b64.
// C/D (16x16 f32, 8 VGPRs): c[v] = D[M = v + 8*(lane>=16)][N = lane&15].
// ---------------------------------------------------------------------------
__global__ __launch_bounds__(NTHREADS)
void gemm128(const float* __restrict__ X, const float* __restrict__ W,
             const float* __restrict__ bias, float* __restrict__ Y, int rows)
{
    __shared__ v2f sW[64 * FDIM];   // 64 KB: pair p holds (W[2p][n], W[2p+1][n])

    const int tid  = threadIdx.x;
    const int wave = tid >> 5;
    const int l    = tid & 31;
    const int h    = l >> 4;        // half-wave
    const int li   = l & 15;

    for (int idx = tid; idx < 64 * FDIM; idx += NTHREADS) {
        const int p = idx >> 7, n = idx & (FDIM - 1);
        v2f w; w.x = W[(2 * p) * FDIM + n]; w.y = W[(2 * p + 1) * FDIM + n];
        sW[idx] = w;
    }
    __syncthreads();

    const int tileRow = blockIdx.x * TILE_ROWS + wave * 16;
    int ar = tileRow + li;                       // A-row this lane feeds
    if (ar > rows - 1) ar = rows - 1;            // clamp; garbage rows never stored
    const float* Xr = X + (size_t)ar * FDIM + 2 * h;

    v8f acc[8] = {};
    for (int kk = 0; kk < KSTEPS; ++kk) {
        const v2f a = *(const v2f*)(Xr + 4 * kk);          // A[M=li][4kk+2h .. +1]
        const v2f* bp = &sW[(2 * kk + h) * FDIM + li];
        #pragma unroll
        for (int j = 0; j < 8; ++j) {
            const v2f b = bp[16 * j];                      // W[k..k+1][16j+li]
            acc[j] = __builtin_amdgcn_wmma_f32_16x16x4_f32(
                false, a, false, b, (short)0, acc[j], false, false);
        }
    }

    #pragma unroll
    for (int j = 0; j < 8; ++j) {
        const int col = 16 * j + li;
        const float bj = bias[col];
        #pragma unroll
        for (int v = 0; v < 8; ++v) {
            const int r = tileRow + v + 8 * h;
            if (r < rows) Y[(size_t)r * FDIM + col] = acc[j][v] + bj;
        }
    }
}

// ---------------------------------------------------------------------------
// Edge GEMM, fused: U = Xe@W_ee + b + e_src[src] + e_dst[dst];  store U;
// gate = silu(U); atomic num += n_dst[dst]*gate, den += gate;
// BatchNorm column sums of U accumulated (LDS reduce, then global atomics).
// ---------------------------------------------------------------------------
__global__ __launch_bounds__(NTHREADS)
void edge_gemm(const float* __restrict__ Xe, const float* __restrict__ W,
               const float* __restrict__ bias,
               const float* __restrict__ e_src, const float* __restrict__ e_dst,
               const float* __restrict__ n_dst,
               const int* __restrict__ src, const int* __restrict__ dst,
               float* __restrict__ edge_upd,
               float* __restrict__ num, float* __restrict__ den,
               float* __restrict__ esum, float* __restrict__ esumsq,
               int rows)
{
    __shared__ v2f sW[64 * FDIM];   // reused as reduction buffer after GEMM

    const int tid  = threadIdx.x;
    const int wave = tid >> 5;
    const int l    = tid & 31;
    const int h    = l >> 4;
    const int li   = l & 15;

    for (int idx = tid; idx < 64 * FDIM; idx += NTHREADS) {
        const int p = idx >> 7, n = idx & (FDIM - 1);
        v2f w; w.x = W[(2 * p) * FDIM + n]; w.y = W[(2 * p + 1) * FDIM + n];
        sW[idx] = w;
    }
    __syncthreads();

    const int tileRow = blockIdx.x * TILE_ROWS + wave * 16;
    int ar = tileRow + li;
    if (ar > rows - 1) ar = rows - 1;
    const float* Xr = Xe + (size_t)ar * FDIM + 2 * h;

    v8f acc[8] = {};
    for (int kk = 0; kk < KSTEPS; ++kk) {
        const v2f a = *(const v2f*)(Xr + 4 * kk);
        const v2f* bp = &sW[(2 * kk + h) * FDIM + li];
        #pragma unroll
        for (int j = 0; j < 8; ++j) {
            const v2f b = bp[16 * j];
            acc[j] = __builtin_amdgcn_wmma_f32_16x16x4_f32(
                false, a, false, b, (short)0, acc[j], false, false);
        }
    }

    // per-lane row metadata for the 8 output rows this lane stores
    int  siv[8], div_[8];
    bool val[8];
    #pragma unroll
    for (int v = 0; v < 8; ++v) {
        const int r = tileRow + v + 8 * h;
        val[v] = (r < rows);
        const int rc = val[v] ? r : 0;
        siv[v] = src[rc];
        div_[v] = dst[rc];
    }

    float lsum[8], lsq[8];
    #pragma unroll
    for (int j = 0; j < 8; ++j) {
        const int col = 16 * j + li;
        const float bj = bias[col];
        lsum[j] = 0.0f; lsq[j] = 0.0f;
        #pragma unroll
        for (int v = 0; v < 8; ++v) {
            if (val[v]) {
                const int r = tileRow + v + 8 * h;
                const size_t so = (size_t)siv[v] * FDIM + col;
                const size_t dofs = (size_t)div_[v] * FDIM + col;
                const float u = acc[j][v] + bj + e_src[so] + e_dst[dofs];
                edge_upd[(size_t)r * FDIM + col] = u;
                const float g = silu_f(u);
                atomicAdd(&num[dofs], n_dst[dofs] * g);
                atomicAdd(&den[dofs], g);
                lsum[j] += u;
                lsq[j]  += u * u;
            }
        }
    }

    // block-level BN-stat reduction (reuse sW space)
    __syncthreads();
    float* red = (float*)sW;          // red[0..127]=sum, red[128..255]=sumsq
    red[tid] = 0.0f;
    __syncthreads();
    #pragma unroll
    for (int j = 0; j < 8; ++j) {
        const int col = 16 * j + li;
        atomicAdd(&red[col], lsum[j]);          // ds_add_f32
        atomicAdd(&red[FDIM + col], lsq[j]);
    }
    __syncthreads();
    if (tid < FDIM) atomicAdd(&esum[tid], red[tid]);
    else            atomicAdd(&esumsq[tid - FDIM], red[tid]);
}

// ---------------------------------------------------------------------------
// node_update = ns_b + num/(den+1e-6); accumulate node BN sums.
// Each block: 64 rows x 128 cols.
// ---------------------------------------------------------------------------
__global__ __launch_bounds__(NTHREADS)
void node_combine(const float* __restrict__ ns_b, const float* __restrict__ num,
                  const float* __restrict__ den, float* __restrict__ node_upd,
                  float* __restrict__ nsum, float* __restrict__ nsumsq, int n)
{
    __shared__ float sS[NTHREADS], sQ[NTHREADS];
    const int t = threadIdx.x;
    const int col = t & 127, rh = t >> 7;
    const int base = blockIdx.x * 64;
    float ls = 0.0f, lq = 0.0f;
    for (int rr = 0; rr < 32; ++rr) {
        const int row = base + rr * 2 + rh;
        if (row < n) {
            const size_t i = (size_t)row * FDIM + col;
            const float gte = num[i] / (den[i] + 1e-6f);
            const float u = ns_b[i] + gte;
            node_upd[i] = u;
            ls += u; lq += u * u;
        }
    }
    sS[t] = ls; sQ[t] = lq;
    __syncthreads();
    if (t < FDIM) {
        atomicAdd(&nsum[t],   sS[t] + sS[t + FDIM]);
        atomicAdd(&nsumsq[t], sQ[t] + sQ[t + FDIM]);
    }
}

__global__ void finalize_stats(const float* __restrict__ esum, const float* __restrict__ esumsq,
                               const float* __restrict__ nsum, const float* __restrict__ nsumsq,
                               const float* __restrict__ g_e, const float* __restrict__ be_e,
                               const float* __restrict__ g_n, const float* __restrict__ be_n,
                               float* __restrict__ escale, float* __restrict__ eshift,
                               float* __restrict__ nscale, float* __restrict__ nshift,
                               int ecount, int ncount)
{
    const int c = threadIdx.x;
    if (c < FDIM) {
        const float me = esum[c] / (float)ecount;
        const float ve = esumsq[c] / (float)ecount - me * me;
        const float se = g_e[c] * rsqrtf(ve + 1e-5f);
        escale[c] = se; eshift[c] = be_e[c] - me * se;
        const float mn = nsum[c] / (float)ncount;
        const float vn = nsumsq[c] / (float)ncount - mn * mn;
        const float sn = g_n[c] * rsqrtf(vn + 1e-5f);
        nscale[c] = sn; nshift[c] = be_n[c] - mn * sn;
    }
}

__global__ __launch_bounds__(NTHREADS)
void bn_silu_res(const float* __restrict__ u, const float* __restrict__ resid,
                 const float* __restrict__ scale, const float* __restrict__ shift,
                 float* __restrict__ out, int total)
{
    const int i = blockIdx.x * NTHREADS + threadIdx.x;
    if (i < total) {
        const int c = i & 127;
        const float x = u[i] * scale[c] + shift[c];
        out[i] = silu_f(x) + resid[i];
    }
}

// ---------------------------------------------------------------------------
extern "C" void kernel_launch(void* const* d_in, const int* in_sizes, int n_in,
                              void* d_out, int out_size, void* d_ws, size_t ws_size,
                              hipStream_t stream) {
    (void)n_in; (void)out_size; (void)ws_size;

    const float* node_feats = (const float*)d_in[0];
    const float* edge_feats = (const float*)d_in[1];
    const int*   src  = (const int*)d_in[2];
    const int*   dst  = (const int*)d_in[3];
    const float* W_es = (const float*)d_in[4];
    const float* b_es = (const float*)d_in[5];
    const float* W_ed = (const float*)d_in[6];
    const float* b_ed = (const float*)d_in[7];
    const float* W_ee = (const float*)d_in[8];
    const float* b_ee = (const float*)d_in[9];
    const float* W_nd = (const float*)d_in[10];
    const float* b_nd = (const float*)d_in[11];
    const float* W_ns = (const float*)d_in[12];
    const float* b_ns = (const float*)d_in[13];
    const float* g_e  = (const float*)d_in[14];
    const float* be_e = (const float*)d_in[15];
    const float* g_n  = (const float*)d_in[16];
    const float* be_n = (const float*)d_in[17];

    const int N_ = in_sizes[0] / FDIM;   // 100000
    const int E_ = in_sizes[1] / FDIM;   // 600000
    const size_t NF = (size_t)N_ * FDIM;
    const size_t EF = (size_t)E_ * FDIM;

    // workspace carve-up (floats): 6*NF + EF + 1024  (~615 MB)
    float* ws      = (float*)d_ws;
    float* e_src   = ws;                 // node->edge src projection
    float* e_dst_b = e_src + NF;         // node->edge dst projection
    float* n_dst_b = e_dst_b + NF;       // W_nd projection
    float* ns_b    = n_dst_b + NF;       // W_ns projection
    float* num     = ns_b + NF;          // gated numerator accumulator
    float* den     = num + NF;           // gate denominator accumulator
    float* edge_upd = den + NF;          // [E,F] pre-BN edge update
    float* stats   = edge_upd + EF;
    float* esum = stats, *esumsq = stats + 128, *nsum = stats + 256, *nsumsq = stats + 384;
    float* escale = stats + 512, *eshift = stats + 640, *nscale = stats + 768, *nshift = stats + 896;
    float* node_upd = e_src;             // e_src dead after edge_gemm -> reuse

    float* node_out = (float*)d_out;     // tuple order: (node_out, edge_out)
    float* edge_out = node_out + NF;

    // zero accumulators every call (graph-replay safe)
    hipMemsetAsync(num, 0, 2 * NF * sizeof(float), stream);
    hipMemsetAsync(stats, 0, 512 * sizeof(float), stream);

    const int nb = (N_ + TILE_ROWS - 1) / TILE_ROWS;
    gemm128<<<nb, NTHREADS, 0, stream>>>(node_feats, W_es, b_es, e_src,   N_);
    gemm128<<<nb, NTHREADS, 0, stream>>>(node_feats, W_ed, b_ed, e_dst_b, N_);
    gemm128<<<nb, NTHREADS, 0, stream>>>(node_feats, W_nd, b_nd, n_dst_b, N_);
    gemm128<<<nb, NTHREADS, 0, stream>>>(node_feats, W_ns, b_ns, ns_b,    N_);

    const int eb = (E_ + TILE_ROWS - 1) / TILE_ROWS;
    edge_gemm<<<eb, NTHREADS, 0, stream>>>(edge_feats, W_ee, b_ee,
                                           e_src, e_dst_b, n_dst_b, src, dst,
                                           edge_upd, num, den, esum, esumsq, E_);

    node_combine<<<(N_ + 63) / 64, NTHREADS, 0, stream>>>(ns_b, num, den, node_upd,
                                                          nsum, nsumsq, N_);
    finalize_stats<<<1, FDIM, 0, stream>>>(esum, esumsq, nsum, nsumsq,
                                           g_e, be_e, g_n, be_n,
                                           escale, eshift, nscale, nshift, E_, N_);

    bn_silu_res<<<(int)((EF + NTHREADS - 1) / NTHREADS), NTHREADS, 0, stream>>>(
        edge_upd, edge_feats, escale, eshift, edge_out, (int)EF);
    bn_silu_res<<<(int)((NF + NTHREADS - 1) / NTHREADS), NTHREADS, 0, stream>>>(
        node_upd, node_feats, nscale, nshift, node_out, (int)NF);
}